// SourceDetectLocalize_9242769622019
// MI455X (gfx1250) — compile-verified
//
#include <hip/hip_runtime.h>
#include <hip/hip_bf16.h>
#include <math.h>

typedef __attribute__((ext_vector_type(2))) float v2f;
typedef __attribute__((ext_vector_type(8))) float v8f;
typedef int vi4 __attribute__((vector_size(16)));  // matches builtin's V4i

namespace {
constexpr int   kBT    = 8192;            // nb*nt = 16*512
constexpr int   kK     = 1024;            // nf*nmic = 256*4
constexpr int   kNazi  = 73;
constexpr int   kC     = 2701;            // nele*nazi = 37*73
constexpr int   kCpad  = 2704;            // next multiple of 16
constexpr float kScale = 2.0f / 1024.0f;  // 2/(nmic*nf)

constexpr int BM  = 128;            // block tile M
constexpr int BN  = 128;            // block tile N
constexpr int KC  = 16;             // K chunk staged in LDS (double buffered)
constexpr int LDS_STRIDE = KC + 4;  // 20: 16B-aligned rows, conflict-free reads
constexpr int BUF = BM * LDS_STRIDE;  // floats per buffer per matrix (2560)
}  // namespace

// Async memory->LDS copy of one 16B chunk per lane (ASYNCcnt tracked).
__device__ __forceinline__ void async_copy_b128(const float* g, float* l) {
#if __has_builtin(__builtin_amdgcn_global_load_async_to_lds_b128)
  __builtin_amdgcn_global_load_async_to_lds_b128(
      (__attribute__((address_space(1))) vi4*)(uintptr_t)g,
      (__attribute__((address_space(3))) vi4*)l, 0, 0);
#else
  unsigned loff =
      (unsigned)(uintptr_t)(__attribute__((address_space(3))) float*)l;
  asm volatile("global_load_async_to_lds_b128 %0, %1, off" ::"v"(loff),
               "v"((unsigned long long)(uintptr_t)g)
               : "memory");
#endif
}

__device__ __forceinline__ void wait_async0() {
#if __has_builtin(__builtin_amdgcn_s_wait_asynccnt)
  __builtin_amdgcn_s_wait_asynccnt(0);
#else
  asm volatile("s_wait_asynccnt 0x0" ::: "memory");
#endif
}

__device__ __forceinline__ int imin(int a, int b) { return a < b ? a : b; }

// out[m][n] = scale * sum_k A[m][k] * B[n][k]
// OOB A/B rows clamp their load address (garbage confined to out rows/cols
// >= the valid count, which callers never store/read). Stores guarded by
// Mstore/Nstore. K fixed at kK=1024. grid = (ceil(M/128), ceil(N/128)),
// block = 256 threads = 8 wave32 waves (4 along M x 2 along N); each wave
// computes a 32x64 tile with eight f32 16x16x4 WMMA accumulators.
// Staging uses GLOBAL_LOAD_ASYNC_TO_LDS_B128 with double-buffered LDS: the
// next K-chunk streams in while the current one is consumed by WMMAs.
__global__ __launch_bounds__(256) void wmma_gemm_nt(
    const float* __restrict__ A, const float* __restrict__ B,
    float* __restrict__ out, int Mvalid, int Mstore, int Nvalid, int Nstore,
    int ldo, float scale) {
  __shared__ float As[2 * BUF];
  __shared__ float Bs[2 * BUF];

  const int tid  = threadIdx.x;
  const int lane = tid & 31;
  const int wave = tid >> 5;
  const int wm   = wave & 3;   // 4 waves along M (32 rows each)
  const int wn   = wave >> 2;  // 2 waves along N (64 cols each)
  const int l16  = lane & 15;
  const int lh   = lane >> 4;  // half-wave selector (K phase)

  const int blockM = blockIdx.x * BM;
  const int blockN = blockIdx.y * BN;

  // Per-thread staging assignment: tile is 128 rows x 16 k = 512 b128 chunks
  // per matrix; 2 chunks per thread per matrix. 4 consecutive lanes cover one
  // contiguous 64B row segment (coalesced).
  int          aoff[2];
  const float* gA[2];
  const float* gB[2];
#pragma unroll
  for (int i = 0; i < 2; ++i) {
    int v   = tid + i * 256;
    int row = v >> 2;
    int kq  = (v & 3) << 2;
    aoff[i] = row * LDS_STRIDE + kq;
    gA[i]   = A + (size_t)imin(blockM + row, Mvalid - 1) * kK + kq;
    gB[i]   = B + (size_t)imin(blockN + row, Nvalid - 1) * kK + kq;
  }

  const v8f zero = {0.f, 0.f, 0.f, 0.f, 0.f, 0.f, 0.f, 0.f};
  v8f acc[2][4];
#pragma unroll
  for (int mi = 0; mi < 2; ++mi)
#pragma unroll
    for (int ni = 0; ni < 4; ++ni) acc[mi][ni] = zero;

  const int arow = (wm * 32 + l16) * LDS_STRIDE;
  const int brow = (wn * 64 + l16) * LDS_STRIDE;

  // Prologue: stream in the first K-chunk.
#pragma unroll
  for (int i = 0; i < 2; ++i) {
    async_copy_b128(gA[i], &As[aoff[i]]);
    async_copy_b128(gB[i], &Bs[aoff[i]]);
  }
  wait_async0();
  __syncthreads();

  for (int kt = 0; kt < kK; kt += KC) {
    const int cur = (kt / KC) & 1;
    // Kick off the next chunk into the other buffer while we compute.
    if (kt + KC < kK) {
      const int nxt = cur ^ 1;
#pragma unroll
      for (int i = 0; i < 2; ++i) {
        async_copy_b128(gA[i] + kt + KC, &As[nxt * BUF + aoff[i]]);
        async_copy_b128(gB[i] + kt + KC, &Bs[nxt * BUF + aoff[i]]);
      }
    }

    const float* Arow0 = &As[cur * BUF + arow];
    const float* Arow1 = Arow0 + 16 * LDS_STRIDE;
    const float* Brow0 = &Bs[cur * BUF + brow];
    const float* Brow1 = Brow0 + 16 * LDS_STRIDE;
    const float* Brow2 = Brow0 + 32 * LDS_STRIDE;
    const float* Brow3 = Brow0 + 48 * LDS_STRIDE;

#pragma unroll
    for (int kk = 0; kk < KC; kk += 4) {
      const int ka = kk + 2 * lh;
      // A frag: lane l<16 holds A[M=l][K=ka..ka+1]; lanes 16-31 K=ka+2..ka+3.
      // B frag: lane l<16 holds B[K=ka..ka+1][N=l] = T[col][ka..ka+1].
      v2f a0 = *(const v2f*)(Arow0 + ka);
      v2f a1 = *(const v2f*)(Arow1 + ka);
      v2f b0 = *(const v2f*)(Brow0 + ka);
      v2f b1 = *(const v2f*)(Brow1 + ka);
      v2f b2 = *(const v2f*)(Brow2 + ka);
      v2f b3 = *(const v2f*)(Brow3 + ka);
      acc[0][0] = __builtin_amdgcn_wmma_f32_16x16x4_f32(
          false, a0, false, b0, (short)0, acc[0][0], false, false);
      acc[0][1] = __builtin_amdgcn_wmma_f32_16x16x4_f32(
          false, a0, false, b1, (short)0, acc[0][1], false, false);
      acc[0][2] = __builtin_amdgcn_wmma_f32_16x16x4_f32(
          false, a0, false, b2, (short)0, acc[0][2], false, false);
      acc[0][3] = __builtin_amdgcn_wmma_f32_16x16x4_f32(
          false, a0, false, b3, (short)0, acc[0][3], false, false);
      acc[1][0] = __builtin_amdgcn_wmma_f32_16x16x4_f32(
          false, a1, false, b0, (short)0, acc[1][0], false, false);
      acc[1][1] = __builtin_amdgcn_wmma_f32_16x16x4_f32(
          false, a1, false, b1, (short)0, acc[1][1], false, false);
      acc[1][2] = __builtin_amdgcn_wmma_f32_16x16x4_f32(
          false, a1, false, b2, (short)0, acc[1][2], false, false);
      acc[1][3] = __builtin_amdgcn_wmma_f32_16x16x4_f32(
          false, a1, false, b3, (short)0, acc[1][3], false, false);
    }

    // Own async writes done, then block-wide visibility before buffer swap.
    wait_async0();
    __syncthreads();
  }

  // C/D layout: VGPR r, lanes 0-15 -> M=r, lanes 16-31 -> M=r+8; N=lane%16.
#pragma unroll
  for (int mi = 0; mi < 2; ++mi) {
#pragma unroll
    for (int ni = 0; ni < 4; ++ni) {
#pragma unroll
      for (int r = 0; r < 8; ++r) {
        int row = blockM + wm * 32 + mi * 16 + r + 8 * lh;
        int col = blockN + wn * 64 + ni * 16 + l16;
        if (row < Mstore && col < Nstore)
          out[(size_t)row * ldo + col] = acc[mi][ni][r] * scale;
      }
    }
  }
}

// Iteration 1: argmax over S row, ratio = (S[idx]/scale)/G[idx,idx].
__global__ __launch_bounds__(256) void argmax_iter1(
    const float* __restrict__ S, const float* __restrict__ G,
    const float* __restrict__ ele_c, const float* __restrict__ azi_c,
    float* __restrict__ doas, float* __restrict__ vads,
    int* __restrict__ idx1_out, float* __restrict__ ratio1_out) {
  __shared__ float sval[256];
  __shared__ int   sidx[256];
  const int    bt  = blockIdx.x;
  const float* row = S + (size_t)bt * kC;

  float best = -INFINITY;
  int   bidx = 0x7fffffff;
  for (int c = threadIdx.x; c < kC; c += 256) {
    float v = row[c];
    if (v > best) { best = v; bidx = c; }
  }
  sval[threadIdx.x] = best;
  sidx[threadIdx.x] = bidx;
  __syncthreads();
  for (int s = 128; s > 0; s >>= 1) {
    if (threadIdx.x < s) {
      float v2 = sval[threadIdx.x + s];
      int   i2 = sidx[threadIdx.x + s];
      if (v2 > sval[threadIdx.x] ||
          (v2 == sval[threadIdx.x] && i2 < sidx[threadIdx.x])) {
        sval[threadIdx.x] = v2;
        sidx[threadIdx.x] = i2;
      }
    }
    __syncthreads();
  }
  if (threadIdx.x == 0) {
    int   idx   = sidx[0];
    float num   = sval[0] / kScale;
    float den   = G[(size_t)idx * kCpad + idx];
    float ratio = num / den;
    int   eidx  = idx / kNazi;
    int   aidx  = idx - eidx * kNazi;
    doas[bt * 4 + 0] = ele_c[eidx];  // [bt][d=0][s=0]
    doas[bt * 4 + 2] = azi_c[aidx];  // [bt][d=1][s=0]
    vads[bt * 2 + 0] = ratio;
    idx1_out[bt]     = idx;
    ratio1_out[bt]   = ratio;
  }
}

// Iteration 2: scores2[c] = S[c] - ratio1*scale*G[idx1][c]; argmax; ratio2.
__global__ __launch_bounds__(256) void argmax_iter2(
    const float* __restrict__ S, const float* __restrict__ G,
    const float* __restrict__ ele_c, const float* __restrict__ azi_c,
    const int* __restrict__ idx1, const float* __restrict__ ratio1,
    float* __restrict__ doas, float* __restrict__ vads) {
  __shared__ float sval[256];
  __shared__ int   sidx[256];
  const int    bt   = blockIdx.x;
  const float* row  = S + (size_t)bt * kC;
  const float* grow = G + (size_t)idx1[bt] * kCpad;
  const float  rs   = ratio1[bt] * kScale;

  float best = -INFINITY;
  int   bidx = 0x7fffffff;
  for (int c = threadIdx.x; c < kC; c += 256) {
    float v = row[c] - rs * grow[c];
    if (v > best) { best = v; bidx = c; }
  }
  sval[threadIdx.x] = best;
  sidx[threadIdx.x] = bidx;
  __syncthreads();
  for (int s = 128; s > 0; s >>= 1) {
    if (threadIdx.x < s) {
      float v2 = sval[threadIdx.x + s];
      int   i2 = sidx[threadIdx.x + s];
      if (v2 > sval[threadIdx.x] ||
          (v2 == sval[threadIdx.x] && i2 < sidx[threadIdx.x])) {
        sval[threadIdx.x] = v2;
        sidx[threadIdx.x] = i2;
      }
    }
    __syncthreads();
  }
  if (threadIdx.x == 0) {
    int   idx   = sidx[0];
    float num   = sval[0] / kScale;  // = S[idx]/scale - ratio1*G[idx1,idx]
    float den   = G[(size_t)idx * kCpad + idx];
    float ratio = num / den;
    int   eidx  = idx / kNazi;
    int   aidx  = idx - eidx * kNazi;
    doas[bt * 4 + 1] = ele_c[eidx];  // [bt][d=0][s=1]
    doas[bt * 4 + 3] = azi_c[aidx];  // [bt][d=1][s=1]
    vads[bt * 2 + 1] = ratio;
  }
}

extern "C" void kernel_launch(void* const* d_in, const int* in_sizes, int n_in,
                              void* d_out, int out_size, void* d_ws,
                              size_t ws_size, hipStream_t stream) {
  (void)in_sizes; (void)n_in; (void)out_size; (void)ws_size;
  const float* pred_ipd = (const float*)d_in[0];  // [8192][1024]
  const float* tmpl     = (const float*)d_in[1];  // [2701][1024]
  const float* ele_c    = (const float*)d_in[2];  // [37]
  const float* azi_c    = (const float*)d_in[3];  // [73]

  float* out  = (float*)d_out;
  float* doas = out;                  // [8192][2][2] = 32768
  float* vads = out + 32768;          // [8192][2]    = 16384
  float* ss   = out + 49152;          // [8192][2701] (also serves as S)

  // Workspace: G (2704*2704 f32 ~ 29.2 MB) + idx1 + ratio1.
  float* G      = (float*)d_ws;
  char*  tail   = (char*)d_ws + (size_t)kCpad * kCpad * sizeof(float);
  int*   idx1   = (int*)tail;
  float* ratio1 = (float*)(tail + (size_t)kBT * sizeof(int));

  dim3 blk(256);

  // S = scale * ipd @ tmpl^T  -> written straight into pred_ss.
  wmma_gemm_nt<<<dim3(kBT / BM, (kC + BN - 1) / BN), blk, 0, stream>>>(
      pred_ipd, tmpl, ss, /*Mvalid=*/kBT, /*Mstore=*/kBT,
      /*Nvalid=*/kC, /*Nstore=*/kC, /*ldo=*/kC, kScale);

  // G = tmpl @ tmpl^T (unscaled), padded leading dim.
  wmma_gemm_nt<<<dim3((kCpad + BM - 1) / BM, (kCpad + BN - 1) / BN), blk, 0,
                 stream>>>(tmpl, tmpl, G, /*Mvalid=*/kC, /*Mstore=*/kCpad,
                           /*Nvalid=*/kC, /*Nstore=*/kCpad, /*ldo=*/kCpad,
                           1.0f);

  argmax_iter1<<<kBT, blk, 0, stream>>>(ss, G, ele_c, azi_c, doas, vads, idx1,
                                        ratio1);
  argmax_iter2<<<kBT, blk, 0, stream>>>(ss, G, ele_c, azi_c, idx1, ratio1,
                                        doas, vads);
}